// PpiKkoTwistGNN_20907900797109
// MI455X (gfx1250) — compile-verified
//
#include <hip/hip_runtime.h>

#define H      128
#define NU     20000
#define NT     50000
#define NV     100000
#define LAYERS 8
#define NEDGE  500000
#define EPSLN  1e-5f
#define HH     (H * H)
#define PS128  16384  // packed size for K=128: (K/4)*8*32*2

typedef float v2f __attribute__((ext_vector_type(2)));
typedef float v8f __attribute__((ext_vector_type(8)));

// ---------------------------------------------------------------------------
// WMMA: D = A(16x4 f32) * B(4x16 f32) + C(16x16 f32), wave32.
// ---------------------------------------------------------------------------
__device__ __forceinline__ v8f wmma_f32(v2f a, v2f b, v8f c) {
  return __builtin_amdgcn_wmma_f32_16x16x4_f32(false, a, false, b, (short)0, c,
                                               false, false);
}

// ---------------------------------------------------------------------------
// Fused multi-term GEMM: D[M,128] = sum_i A_i @ W_i (+bias)(+relu).
// W_i are pre-packed in fragment order [K/4][8 ntiles][32 lanes][2], so each
// B-fragment is one coalesced b64 load. One wave per 16-row strip (8 v8f acc).
// ---------------------------------------------------------------------------
__global__ __launch_bounds__(256) void gemm_fused(
    const float* __restrict__ A1, const float* __restrict__ P1,
    const float* __restrict__ A2, const float* __restrict__ P2,
    const float* __restrict__ A3, const float* __restrict__ P3,
    const float* __restrict__ bias, float* __restrict__ D, int M, int K,
    int relu) {
  const int lane  = threadIdx.x & 31;
  const int strip = blockIdx.x * 8 + (threadIdx.x >> 5);
  if (strip * 16 >= M) return;  // wave-uniform: EXEC all-ones for WMMA
  const int m0 = strip * 16;
  // A-frag (ISA 7.12.2): lanes 0-15 hold K={k0,k0+1} for M=lane&15,
  // lanes 16-31 hold K={k0+2,k0+3}.
  const size_t abase = (size_t)(m0 + (lane & 15)) * K + ((lane >> 4) << 1);
  const float* a1p = A1 + abase;
  const float* a2p = A2 ? A2 + abase : nullptr;
  const float* a3p = A3 ? A3 + abase : nullptr;

  v8f acc[8] = {};
  const int K4 = K >> 2;
  for (int ks = 0; ks < K4; ++ks) {
    const int kofs = ks << 2;
    {
      v2f a = *(const v2f*)(a1p + kofs);
      const float* bp = P1 + ((size_t)ks << 9) + (lane << 1);
#pragma unroll
      for (int j = 0; j < 8; ++j)
        acc[j] = wmma_f32(a, *(const v2f*)(bp + (j << 6)), acc[j]);
    }
    if (A2) {
      v2f a = *(const v2f*)(a2p + kofs);
      const float* bp = P2 + ((size_t)ks << 9) + (lane << 1);
#pragma unroll
      for (int j = 0; j < 8; ++j)
        acc[j] = wmma_f32(a, *(const v2f*)(bp + (j << 6)), acc[j]);
    }
    if (A3) {
      v2f a = *(const v2f*)(a3p + kofs);
      const float* bp = P3 + ((size_t)ks << 9) + (lane << 1);
#pragma unroll
      for (int j = 0; j < 8; ++j)
        acc[j] = wmma_f32(a, *(const v2f*)(bp + (j << 6)), acc[j]);
    }
  }

  // C/D layout: VGPR v, lanes 0-15 -> (M=m0+v, N=lane); lanes 16-31 -> M=m0+8+v
#pragma unroll
  for (int j = 0; j < 8; ++j) {
    const int col = j * 16 + (lane & 15);
    const float bb = bias ? bias[col] : 0.0f;
#pragma unroll
    for (int v = 0; v < 8; ++v) {
      const int row = m0 + v + ((lane >> 4) << 3);
      float val = acc[j][v] + bb;
      if (relu) val = fmaxf(val, 0.0f);
      D[(size_t)row * H + col] = val;
    }
  }
}

// Pack a [K,128] weight into fragment order: idx i -> (ks,j,lane), two floats.
__global__ __launch_bounds__(256) void pack_w(const float* __restrict__ W,
                                              float* __restrict__ P, int K) {
  const int i = blockIdx.x * 256 + threadIdx.x;
  if (i >= (K >> 2) * 256) return;
  const int lane = i & 31;
  const int j    = (i >> 5) & 7;
  const int ks   = i >> 8;
  const int col  = j * 16 + (lane & 15);
  const int kb   = ks * 4 + ((lane >> 4) << 1);
  P[i * 2 + 0] = W[(size_t)kb * H + col];
  P[i * 2 + 1] = W[(size_t)(kb + 1) * H + col];
}

// Same, but packs W0 + W3 (fused root-weight sum for the travel update).
__global__ __launch_bounds__(256) void pack_wsum(const float* __restrict__ W0,
                                                 const float* __restrict__ W3,
                                                 float* __restrict__ P) {
  const int i = blockIdx.x * 256 + threadIdx.x;
  if (i >= 32 * 256) return;
  const int lane = i & 31;
  const int j    = (i >> 5) & 7;
  const int ks   = i >> 8;
  const int col  = j * 16 + (lane & 15);
  const int kb   = ks * 4 + ((lane >> 4) << 1);
  const size_t o0 = (size_t)kb * H + col;
  const size_t o1 = (size_t)(kb + 1) * H + col;
  P[i * 2 + 0] = W0[o0] + W3[o0];
  P[i * 2 + 1] = W0[o1] + W3[o1];
}

// ---------------------------------------------------------------------------
// CSR build (once per launch; graph is launch-invariant)
// ---------------------------------------------------------------------------
__global__ __launch_bounds__(256) void count_int(const int* __restrict__ ei,
                                                 int* __restrict__ cnt, int E) {
  const int e = blockIdx.x * 256 + threadIdx.x;
  if (e < E) atomicAdd(&cnt[ei[E + e]], 1);
}

__global__ __launch_bounds__(256) void scan1(const int* __restrict__ cnt,
                                             int* __restrict__ incl,
                                             int* __restrict__ bsum, int n) {
  __shared__ int sh[256];
  const int i = blockIdx.x * 256 + threadIdx.x;
  sh[threadIdx.x] = (i < n) ? cnt[i] : 0;
  __syncthreads();
  for (int off = 1; off < 256; off <<= 1) {
    const int t = (threadIdx.x >= off) ? sh[threadIdx.x - off] : 0;
    __syncthreads();
    sh[threadIdx.x] += t;
    __syncthreads();
  }
  if (i < n) incl[i] = sh[threadIdx.x];
  if (threadIdx.x == 255) bsum[blockIdx.x] = sh[255];
}

__global__ void scan2(int* __restrict__ bsum, int nb) {
  if (threadIdx.x == 0 && blockIdx.x == 0) {
    int run = 0;
    for (int b = 0; b < nb; ++b) {
      const int t = bsum[b];
      bsum[b] = run;
      run += t;
    }
  }
}

__global__ __launch_bounds__(256) void scan3(const int* __restrict__ incl,
                                             const int* __restrict__ bsum,
                                             int* __restrict__ off, int n) {
  const int i = blockIdx.x * 256 + threadIdx.x;
  if (i < n) {
    off[i + 1] = incl[i] + bsum[blockIdx.x];
    if (i == 0) off[0] = 0;
  }
}

__global__ __launch_bounds__(256) void fill_csr(const int* __restrict__ ei,
                                                int* __restrict__ cursor,
                                                int* __restrict__ srcs, int E) {
  const int e = blockIdx.x * 256 + threadIdx.x;
  if (e < E) {
    const int pos = atomicAdd(&cursor[ei[E + e]], 1);
    srcs[pos] = ei[e];
  }
}

__global__ __launch_bounds__(256) void make_inv(const int* __restrict__ cnt,
                                                float* __restrict__ inv,
                                                int n) {
  const int i = blockIdx.x * 256 + threadIdx.x;
  if (i < n) inv[i] = 1.0f / fmaxf((float)cnt[i], 1.0f);
}

// ---------------------------------------------------------------------------
// Pull-style mean aggregation: one wave per destination row, register
// accumulation, no atomics, streaming float4 writes of the mean.
// ---------------------------------------------------------------------------
__global__ __launch_bounds__(256) void gather_mean(
    const float* __restrict__ xsrc, const int* __restrict__ off,
    const int* __restrict__ srcs, const float* __restrict__ inv,
    float* __restrict__ agg, int n) {
  const int lane = threadIdx.x & 31;
  const int row = blockIdx.x * 8 + (threadIdx.x >> 5);
  if (row >= n) return;
  const int beg = off[row], end = off[row + 1];
  float ax = 0.f, ay = 0.f, az = 0.f, aw = 0.f;
  for (int e = beg; e < end; ++e) {
    const float4 v = ((const float4*)(xsrc + (size_t)srcs[e] * H))[lane];
    ax += v.x; ay += v.y; az += v.z; aw += v.w;
  }
  const float s = inv[row];
  float4 o;
  o.x = ax * s; o.y = ay * s; o.z = az * s; o.w = aw * s;
  ((float4*)(agg + (size_t)row * H))[lane] = o;
}

// ---------------------------------------------------------------------------
// LayerNorm over H=128 (wave per row), optional 2 pre-LN biases, ReLU,
// post-ReLU residual (matches reference: x = relu(ln(h+b)) + x).
// ---------------------------------------------------------------------------
__global__ __launch_bounds__(256) void ln_kernel(
    const float* __restrict__ h, const float* __restrict__ b1,
    const float* __restrict__ b2, const float* __restrict__ g,
    const float* __restrict__ be, const float* __restrict__ res,
    float* __restrict__ out, int M, int relu) {
  const int lane = threadIdx.x & 31;
  const int row = blockIdx.x * 8 + (threadIdx.x >> 5);
  if (row >= M) return;
  const int c = lane * 4;
  float4 v = ((const float4*)(h + (size_t)row * H))[lane];
  if (b1) { v.x += b1[c]; v.y += b1[c + 1]; v.z += b1[c + 2]; v.w += b1[c + 3]; }
  if (b2) { v.x += b2[c]; v.y += b2[c + 1]; v.z += b2[c + 2]; v.w += b2[c + 3]; }
  float s = v.x + v.y + v.z + v.w;
#pragma unroll
  for (int off = 16; off; off >>= 1) s += __shfl_xor(s, off, 32);
  const float mean = s * (1.0f / H);
  const float dx = v.x - mean, dy = v.y - mean, dz = v.z - mean, dw = v.w - mean;
  float q = dx * dx + dy * dy + dz * dz + dw * dw;
#pragma unroll
  for (int off = 16; off; off >>= 1) q += __shfl_xor(q, off, 32);
  const float rstd = rsqrtf(q * (1.0f / H) + EPSLN);
  float4 y;
  y.x = dx * rstd * g[c] + be[c];
  y.y = dy * rstd * g[c + 1] + be[c + 1];
  y.z = dz * rstd * g[c + 2] + be[c + 2];
  y.w = dw * rstd * g[c + 3] + be[c + 3];
  if (relu) {
    y.x = fmaxf(y.x, 0.f); y.y = fmaxf(y.y, 0.f);
    y.z = fmaxf(y.z, 0.f); y.w = fmaxf(y.w, 0.f);
  }
  if (res) {
    const float4 rr = ((const float4*)(res + (size_t)row * H))[lane];
    y.x += rr.x; y.y += rr.y; y.z += rr.z; y.w += rr.w;
  }
  ((float4*)(out + (size_t)row * H))[lane] = y;
}

__global__ __launch_bounds__(256) void final_dot(
    const float* __restrict__ z, const float* __restrict__ w,
    const float* __restrict__ b, float* __restrict__ out, int M) {
  const int lane = threadIdx.x & 31;
  const int row = blockIdx.x * 8 + (threadIdx.x >> 5);
  if (row >= M) return;
  const float4 zv = ((const float4*)(z + (size_t)row * H))[lane];
  const float4 wv = ((const float4*)w)[lane];
  float s = zv.x * wv.x + zv.y * wv.y + zv.z * wv.z + zv.w * wv.w;
#pragma unroll
  for (int off = 16; off; off >>= 1) s += __shfl_xor(s, off, 32);
  if (lane == 0) out[row] = s + b[0];
}

// ---------------------------------------------------------------------------
static void build_csr(const int* ei, int n_dst, int* cnt, int* incl, int* bsum,
                      int* off, int* cursor, int* srcs, float* inv,
                      hipStream_t stream) {
  const int nb = (n_dst + 255) / 256;
  hipMemsetAsync(cnt, 0, (size_t)n_dst * sizeof(int), stream);
  count_int<<<(NEDGE + 255) / 256, 256, 0, stream>>>(ei, cnt, NEDGE);
  scan1<<<nb, 256, 0, stream>>>(cnt, incl, bsum, n_dst);
  scan2<<<1, 32, 0, stream>>>(bsum, nb);
  scan3<<<nb, 256, 0, stream>>>(incl, bsum, off, n_dst);
  hipMemcpyAsync(cursor, off, (size_t)n_dst * sizeof(int),
                 hipMemcpyDeviceToDevice, stream);
  fill_csr<<<(NEDGE + 255) / 256, 256, 0, stream>>>(ei, cursor, srcs, NEDGE);
  make_inv<<<nb, 256, 0, stream>>>(cnt, inv, n_dst);
}

extern "C" void kernel_launch(void* const* d_in, const int* in_sizes, int n_in,
                              void* d_out, int out_size, void* d_ws,
                              size_t ws_size, hipStream_t stream) {
  (void)in_sizes; (void)n_in; (void)out_size; (void)ws_size;
  const float* x_user      = (const float*)d_in[0];
  const float* x_travel    = (const float*)d_in[1];
  const float* x_visit     = (const float*)d_in[2];
  const float* W_in_user   = (const float*)d_in[3];
  const float* b_in_user   = (const float*)d_in[4];
  const float* W_in_travel = (const float*)d_in[5];
  const float* b_in_travel = (const float*)d_in[6];
  const float* Wl          = (const float*)d_in[7];
  const float* bl          = (const float*)d_in[8];
  const float* Wr          = (const float*)d_in[9];
  const float* ln_g        = (const float*)d_in[10];
  const float* ln_b        = (const float*)d_in[11];
  // d_in[12..19] expert heads: dead code. d_in[20..24] Q/K path: dead
  // (softmax over a single key is identically 1).
  const float* Wv     = (const float*)d_in[25];
  const float* bvv    = (const float*)d_in[26];
  const float* Wo     = (const float*)d_in[27];
  const float* bo     = (const float*)d_in[28];
  const float* f_ln_g = (const float*)d_in[29];
  const float* f_ln_b = (const float*)d_in[30];
  const float* f_W1   = (const float*)d_in[31];
  const float* f_b1   = (const float*)d_in[32];
  const float* f_W2   = (const float*)d_in[33];
  const float* f_b2   = (const float*)d_in[34];
  const int* ei_ut = (const int*)d_in[35];
  const int* ei_tu = (const int*)d_in[36];
  const int* ei_tv = (const int*)d_in[37];
  const int* ei_vt = (const int*)d_in[38];

  // ---- workspace carve-up (16B-aligned float4 regions first) ----
  float* ws = (float*)d_ws;
  float* xu     = ws; ws += (size_t)NU * H;
  float* xt     = ws; ws += (size_t)NT * H;
  float* xv     = ws; ws += (size_t)NV * H;
  float* hu     = ws; ws += (size_t)NU * H;
  float* ht     = ws; ws += (size_t)NT * H;
  float* hv     = ws; ws += (size_t)NV * H;
  float* agg_tu = ws; ws += (size_t)NU * H;
  float* agg_ut = ws; ws += (size_t)NT * H;
  float* agg_vt = ws; ws += (size_t)NT * H;
  float* agg_tv = ws; ws += (size_t)NV * H;
  float* inv_tu = ws; ws += NU;
  float* inv_ut = ws; ws += NT;
  float* inv_vt = ws; ws += NT;
  float* inv_tv = ws; ws += NV;
  float* Pin_u  = ws; ws += 8192;          // K=64 packed
  float* Pin_t  = ws; ws += 4096;          // K=32 packed
  float* PWl    = ws; ws += 32 * PS128;    // Wl[l,t]
  float* PWr1   = ws; ws += 8 * PS128;     // Wr[l,1]
  float* PWr2   = ws; ws += 8 * PS128;     // Wr[l,2]
  float* PWrs   = ws; ws += 8 * PS128;     // Wr[l,0]+Wr[l,3]
  float* PWv    = ws; ws += PS128;
  float* PWo    = ws; ws += PS128;
  float* PW1    = ws; ws += PS128;
  int* iw = (int*)ws;
  int* srcs_tu = iw; iw += NEDGE;
  int* srcs_ut = iw; iw += NEDGE;
  int* srcs_vt = iw; iw += NEDGE;
  int* srcs_tv = iw; iw += NEDGE;
  int* off_tu  = iw; iw += NU + 1;
  int* off_ut  = iw; iw += NT + 1;
  int* off_vt  = iw; iw += NT + 1;
  int* off_tv  = iw; iw += NV + 1;
  int* cnt_s   = iw; iw += NV;   // shared scratch (stream-ordered reuse)
  int* incl_s  = iw; iw += NV;
  int* curs_s  = iw; iw += NV;
  int* bsum_s  = iw; iw += 512;

  const dim3 blk(256);
  const int gU = (NU / 16 + 7) / 8, gT = (NT / 16 + 7) / 8, gV = (NV / 16 + 7) / 8;
  const int rU = (NU + 7) / 8, rT = (NT + 7) / 8, rV = (NV + 7) / 8;

  // ---- CSR per edge type (layer-invariant) ----
  build_csr(ei_tu, NU, cnt_s, incl_s, bsum_s, off_tu, curs_s, srcs_tu, inv_tu, stream);
  build_csr(ei_ut, NT, cnt_s, incl_s, bsum_s, off_ut, curs_s, srcs_ut, inv_ut, stream);
  build_csr(ei_vt, NT, cnt_s, incl_s, bsum_s, off_vt, curs_s, srcs_vt, inv_vt, stream);
  build_csr(ei_tv, NV, cnt_s, incl_s, bsum_s, off_tv, curs_s, srcs_tv, inv_tv, stream);

  // ---- pack all GEMM weights into WMMA-fragment order ----
  pack_w<<<16, blk, 0, stream>>>(W_in_user, Pin_u, 64);
  pack_w<<<8, blk, 0, stream>>>(W_in_travel, Pin_t, 32);
  for (int l = 0; l < LAYERS; ++l) {
    for (int t = 0; t < 4; ++t)
      pack_w<<<32, blk, 0, stream>>>(Wl + (size_t)(l * 4 + t) * HH,
                                     PWl + (size_t)(l * 4 + t) * PS128, H);
    pack_w<<<32, blk, 0, stream>>>(Wr + (size_t)(l * 4 + 1) * HH,
                                   PWr1 + (size_t)l * PS128, H);
    pack_w<<<32, blk, 0, stream>>>(Wr + (size_t)(l * 4 + 2) * HH,
                                   PWr2 + (size_t)l * PS128, H);
    pack_wsum<<<32, blk, 0, stream>>>(Wr + (size_t)(l * 4 + 0) * HH,
                                      Wr + (size_t)(l * 4 + 3) * HH,
                                      PWrs + (size_t)l * PS128);
  }
  pack_w<<<32, blk, 0, stream>>>(Wv, PWv, H);
  pack_w<<<32, blk, 0, stream>>>(Wo, PWo, H);
  pack_w<<<32, blk, 0, stream>>>(f_W1, PW1, H);

  // ---- input projections ----
  gemm_fused<<<gU, blk, 0, stream>>>(x_user, Pin_u, nullptr, nullptr, nullptr,
                                     nullptr, b_in_user, xu, NU, 64, 0);
  gemm_fused<<<gT, blk, 0, stream>>>(x_travel, Pin_t, nullptr, nullptr, nullptr,
                                     nullptr, b_in_travel, xt, NT, 32, 0);
  hipMemcpyAsync(xv, x_visit, (size_t)NV * H * sizeof(float),
                 hipMemcpyDeviceToDevice, stream);

  // ---- 8 hetero-SAGE layers ----
  for (int l = 0; l < LAYERS; ++l) {
    const float* bl_l = bl + (size_t)l * 4 * H;

    gather_mean<<<rU, blk, 0, stream>>>(xt, off_tu, srcs_tu, inv_tu, agg_tu, NU);
    gather_mean<<<rT, blk, 0, stream>>>(xu, off_ut, srcs_ut, inv_ut, agg_ut, NT);
    gather_mean<<<rT, blk, 0, stream>>>(xv, off_vt, srcs_vt, inv_vt, agg_vt, NT);
    gather_mean<<<rV, blk, 0, stream>>>(xt, off_tv, srcs_tv, inv_tv, agg_tv, NV);

    // hu = mean_tu @ Wl[1] + xu @ Wr[1]
    gemm_fused<<<gU, blk, 0, stream>>>(agg_tu, PWl + (size_t)(l * 4 + 1) * PS128,
                                       xu, PWr1 + (size_t)l * PS128, nullptr,
                                       nullptr, nullptr, hu, NU, H, 0);
    // ht = mean_ut @ Wl[0] + mean_vt @ Wl[3] + xt @ (Wr[0]+Wr[3])
    gemm_fused<<<gT, blk, 0, stream>>>(agg_ut, PWl + (size_t)(l * 4 + 0) * PS128,
                                       agg_vt, PWl + (size_t)(l * 4 + 3) * PS128,
                                       xt, PWrs + (size_t)l * PS128, nullptr,
                                       ht, NT, H, 0);
    // hv = mean_tv @ Wl[2] + xv @ Wr[2]
    gemm_fused<<<gV, blk, 0, stream>>>(agg_tv, PWl + (size_t)(l * 4 + 2) * PS128,
                                       xv, PWr2 + (size_t)l * PS128, nullptr,
                                       nullptr, nullptr, hv, NV, H, 0);

    ln_kernel<<<rU, blk, 0, stream>>>(hu, bl_l + 1 * H, nullptr,
                                      ln_g + (size_t)(l * 3 + 0) * H,
                                      ln_b + (size_t)(l * 3 + 0) * H, xu, xu, NU, 1);
    ln_kernel<<<rT, blk, 0, stream>>>(ht, bl_l + 0 * H, bl_l + 3 * H,
                                      ln_g + (size_t)(l * 3 + 1) * H,
                                      ln_b + (size_t)(l * 3 + 1) * H, xt, xt, NT, 1);
    ln_kernel<<<rV, blk, 0, stream>>>(hv, bl_l + 2 * H, nullptr,
                                      ln_g + (size_t)(l * 3 + 2) * H,
                                      ln_b + (size_t)(l * 3 + 2) * H, xv, xv, NV, 1);
  }

  // ---- head: attn == (xv@Wv+bv)@Wo+bo; then ln -> fc1+relu -> dot ----
  gemm_fused<<<gV, blk, 0, stream>>>(xv, PWv, nullptr, nullptr, nullptr,
                                     nullptr, bvv, hv, NV, H, 0);
  gemm_fused<<<gV, blk, 0, stream>>>(hv, PWo, nullptr, nullptr, nullptr,
                                     nullptr, bo, agg_tv, NV, H, 0);
  ln_kernel<<<rV, blk, 0, stream>>>(agg_tv, nullptr, nullptr, f_ln_g, f_ln_b,
                                    nullptr, hv, NV, 0);
  gemm_fused<<<gV, blk, 0, stream>>>(hv, PW1, nullptr, nullptr, nullptr,
                                     nullptr, f_b1, agg_tv, NV, H, 1);
  final_dot<<<rV, blk, 0, stream>>>(agg_tv, f_W2, f_b2, (float*)d_out, NV);
}